// LinearAttnFirst_37752762532669
// MI455X (gfx1250) — compile-verified
//
#include <hip/hip_runtime.h>
#include <math.h>

#define DIM      128
#define NHEADS   16
#define DPH      8
#define QKV_N    384   // 3*DIM

typedef float v2f __attribute__((ext_vector_type(2)));
typedef float v8f __attribute__((ext_vector_type(8)));

// ---------------------------------------------------------------------------
// fp32 WMMA 16x16x4 (CDNA5 V_WMMA_F32_16X16X4_F32). 8-arg form:
// (neg_a, A, neg_b, B, c_mod, C, reuse_a, reuse_b)
// ---------------------------------------------------------------------------
__device__ __forceinline__ v8f wmma_f32_16x16x4(v2f a, v2f b, v8f c) {
    return __builtin_amdgcn_wmma_f32_16x16x4_f32(
        false, a, false, b, (short)0, c, false, false);
}

// ---------------------------------------------------------------------------
// Pass 0: per-atom LayerNorm stats (mu, rsqrt(var+eps)). One wave per atom.
// ---------------------------------------------------------------------------
__global__ void ln_stats_kernel(const float* __restrict__ nf,
                                float* __restrict__ stats, int natoms) {
    const int atom = blockIdx.x * 8 + (threadIdx.x >> 5);
    const int lane = threadIdx.x & 31;
    if (atom >= natoms) return;
    const float4 v = ((const float4*)(nf + (size_t)atom * DIM))[lane];
    float s  = v.x + v.y + v.z + v.w;
    float s2 = v.x*v.x + v.y*v.y + v.z*v.z + v.w*v.w;
    #pragma unroll
    for (int m = 1; m < 32; m <<= 1) {
        s  += __shfl_xor(s,  m, 32);
        s2 += __shfl_xor(s2, m, 32);
    }
    if (lane == 0) {
        const float mu  = s * (1.0f / DIM);
        const float var = s2 * (1.0f / DIM) - mu * mu;
        stats[2 * atom + 0] = mu;
        stats[2 * atom + 1] = rsqrtf(var + 1e-5f);
    }
}

// ---------------------------------------------------------------------------
// Pass 1: fused LayerNorm + QKV projection via f32 WMMA.
// grid = (M/16, 3), block = 256 (8 waves). Wave w -> N-tile blockIdx.y*8+w.
// A-operand (16x4 LN(h) tile) built on the fly from node_feat + stats.
// ---------------------------------------------------------------------------
__global__ void qkv_gemm_kernel(const float* __restrict__ nf,
                                const float* __restrict__ stats,
                                const float* __restrict__ ln_g,
                                const float* __restrict__ ln_b,
                                const float* __restrict__ w,    // DIM x QKV_N
                                const float* __restrict__ bias, // QKV_N
                                float* __restrict__ out,        // natoms x QKV_N
                                int natoms) {
    const int lane = threadIdx.x & 31;
    const int wid  = threadIdx.x >> 5;
    const int half = lane >> 4;   // 0: K pair {0,1}, 1: K pair {2,3}
    const int l16  = lane & 15;

    const int mbase = blockIdx.x * 16;
    const int nbase = (blockIdx.y * 8 + wid) * 16;
    const int ncol  = nbase + l16;

    int arow = mbase + l16;
    if (arow >= natoms) arow = natoms - 1;      // clamp; store is predicated
    const float mu = stats[2 * arow + 0];
    const float rs = stats[2 * arow + 1];
    const float* __restrict__ ap = nf + (size_t)arow * DIM;

    v8f acc = {};
    #pragma unroll 4
    for (int k0 = 0; k0 < DIM; k0 += 4) {
        const int kk = k0 + 2 * half;
        v2f a, b;
        a.x = (ap[kk]     - mu) * rs * ln_g[kk]     + ln_b[kk];
        a.y = (ap[kk + 1] - mu) * rs * ln_g[kk + 1] + ln_b[kk + 1];
        b.x = w[(size_t)kk       * QKV_N + ncol];
        b.y = w[(size_t)(kk + 1) * QKV_N + ncol];
        acc = wmma_f32_16x16x4(a, b, acc);
    }

    const float bv = bias[ncol];
    #pragma unroll
    for (int r = 0; r < 8; ++r) {
        const int rowm = mbase + r + half * 8;
        if (rowm < natoms)
            out[(size_t)rowm * QKV_N + ncol] = acc[r] + bv;
    }
}

// ---------------------------------------------------------------------------
// Zero-fill helper (d_out / workspace are poisoned, atomics need zeros).
// ---------------------------------------------------------------------------
__global__ void zero_kernel(float* __restrict__ p, size_t n) {
    size_t i = (size_t)blockIdx.x * blockDim.x + threadIdx.x;
    const size_t stride = (size_t)gridDim.x * blockDim.x;
    for (; i < n; i += stride) p[i] = 0.0f;
}

// ---------------------------------------------------------------------------
// Pass 2: edge phase. 128 threads per edge; thread t -> (head t/8, dim t%8).
// Per-head q.k dot via __shfl_xor over the 8-lane head group (wave32).
// m_feat scatter-add -> workspace; m_vec scatter-add -> out_vec directly.
// ---------------------------------------------------------------------------
__global__ void edge_kernel(const float* __restrict__ qkv,      // natoms x 384
                            const float* __restrict__ edge_feat,// E x 128
                            const float* __restrict__ edge_vec, // E x 3
                            const float* __restrict__ radial,   // E
                            const int* __restrict__ row,
                            const int* __restrict__ col,
                            float* __restrict__ m_feat,         // natoms x 128
                            float* __restrict__ out_vec,        // natoms x 3 x 128
                            int nedges) {
    const long long gid = (long long)blockIdx.x * blockDim.x + threadIdx.x;
    const int e = (int)(gid >> 7);
    if (e >= nedges) return;
    const int t = (int)(gid & 127);
    const int h = t >> 3;
    const int d = t & 7;

    const int r = row[e];
    const int c = col[e];

    // qkv layout: (atom, head, 24) with q=[0:8), k=[8:16), v=[16:24)
    const float q = qkv[(size_t)r * QKV_N + h * 24 + d];
    const float k = qkv[(size_t)c * QKV_N + h * 24 + 8  + d];
    const float v = qkv[(size_t)c * QKV_N + h * 24 + 16 + d];

    // head-local dot product over 8 lanes (groups are lane-aligned)
    float p = q * k;
    p += __shfl_xor(p, 1, 32);
    p += __shfl_xor(p, 2, 32);
    p += __shfl_xor(p, 4, 32);

    // exact GELU: 0.5*x*(1+erf(x/sqrt(2))), scaled by radial
    const float attn = 0.5f * p * (1.0f + erff(p * 0.70710678118654752440f))
                     * radial[e];

    const float mf = v * edge_feat[(size_t)e * DIM + t] * attn;
    atomicAdd(&m_feat[(size_t)r * DIM + t], mf);

    const float ev0 = edge_vec[(size_t)e * 3 + 0];
    const float ev1 = edge_vec[(size_t)e * 3 + 1];
    const float ev2 = edge_vec[(size_t)e * 3 + 2];
    float* vb = out_vec + (size_t)r * (3 * DIM) + t;
    atomicAdd(vb,           v * ev0);
    atomicAdd(vb + DIM,     v * ev1);
    atomicAdd(vb + 2 * DIM, v * ev2);
}

// ---------------------------------------------------------------------------
// Pass 3: output projection m_feat @ out_w + out_b via f32 WMMA.
// grid = (M/16), block = 256 (8 waves), wave w -> N-tile w (128/16 = 8).
// ---------------------------------------------------------------------------
__global__ void out_gemm_kernel(const float* __restrict__ mf,   // natoms x DIM
                                const float* __restrict__ w,    // DIM x DIM
                                const float* __restrict__ bias, // DIM
                                float* __restrict__ out,        // natoms x DIM
                                int natoms) {
    const int lane = threadIdx.x & 31;
    const int wid  = threadIdx.x >> 5;
    const int half = lane >> 4;
    const int l16  = lane & 15;

    const int mbase = blockIdx.x * 16;
    const int nbase = wid * 16;
    const int ncol  = nbase + l16;

    int arow = mbase + l16;
    if (arow >= natoms) arow = natoms - 1;
    const float* __restrict__ ap = mf + (size_t)arow * DIM;

    v8f acc = {};
    #pragma unroll 4
    for (int k0 = 0; k0 < DIM; k0 += 4) {
        const int kk = k0 + 2 * half;
        v2f a, b;
        a.x = ap[kk];
        a.y = ap[kk + 1];
        b.x = w[(size_t)kk       * DIM + ncol];
        b.y = w[(size_t)(kk + 1) * DIM + ncol];
        acc = wmma_f32_16x16x4(a, b, acc);
    }

    const float bv = bias[ncol];
    #pragma unroll
    for (int r = 0; r < 8; ++r) {
        const int rowm = mbase + r + half * 8;
        if (rowm < natoms)
            out[(size_t)rowm * DIM + ncol] = acc[r] + bv;
    }
}

// ---------------------------------------------------------------------------
extern "C" void kernel_launch(void* const* d_in, const int* in_sizes, int n_in,
                              void* d_out, int out_size, void* d_ws, size_t ws_size,
                              hipStream_t stream) {
    const float* nf   = (const float*)d_in[0];   // node_feat (natoms,128)
    const float* ef   = (const float*)d_in[1];   // edge_feat (E,16,8)
    const float* evec = (const float*)d_in[2];   // edge_vec  (E,3,1,1)
    const float* rad  = (const float*)d_in[3];   // radial    (E,)
    const float* qkvw = (const float*)d_in[4];   // (128,384)
    const float* qkvb = (const float*)d_in[5];   // (384,)
    const float* outw = (const float*)d_in[6];   // (128,128)
    const float* outb = (const float*)d_in[7];   // (128,)
    const float* lng  = (const float*)d_in[8];   // (128,)
    const float* lnb  = (const float*)d_in[9];   // (128,)
    const int*   row  = (const int*)d_in[10];
    const int*   col  = (const int*)d_in[11];

    const int natoms = in_sizes[0] / DIM;
    const int nedges = in_sizes[3];

    // workspace layout: stats | qkv | m_feat
    float* stats = (float*)d_ws;                             // 2*natoms
    float* qkv   = stats + (size_t)2 * natoms;               // natoms*384
    float* mfeat = qkv + (size_t)natoms * QKV_N;             // natoms*128

    float* out_feat = (float*)d_out;                         // natoms*128
    float* out_vec  = out_feat + (size_t)natoms * DIM;       // natoms*3*128

    // 0) LN stats
    ln_stats_kernel<<<(natoms + 7) / 8, 256, 0, stream>>>(nf, stats, natoms);

    // 1) fused LN + QKV GEMM (WMMA f32)
    dim3 g1((natoms + 15) / 16, 3);
    qkv_gemm_kernel<<<g1, 256, 0, stream>>>(nf, stats, lng, lnb, qkvw, qkvb,
                                            qkv, natoms);

    // zero accumulators (m_feat workspace + vec part of d_out)
    const size_t zc1 = (size_t)natoms * DIM;
    const size_t zc2 = (size_t)natoms * 3 * DIM;
    zero_kernel<<<2048, 256, 0, stream>>>(mfeat, zc1);
    zero_kernel<<<4096, 256, 0, stream>>>(out_vec, zc2);

    // 2) edge phase: attention + segment-sum scatters
    const long long tot = (long long)nedges * DIM;
    const unsigned eblocks = (unsigned)((tot + 255) / 256);
    edge_kernel<<<eblocks, 256, 0, stream>>>(qkv, ef, evec, rad, row, col,
                                             mfeat, out_vec, nedges);

    // 3) output projection (WMMA f32) -> delta_node_feat
    dim3 g2((natoms + 15) / 16, 1);
    out_gemm_kernel<<<g2, 256, 0, stream>>>(mfeat, outw, outb, out_feat, natoms);
}